// CrossModalGNN_81303730913367
// MI455X (gfx1250) — compile-verified
//
#include <hip/hip_runtime.h>
#include <math.h>

// Problem constants (match reference)
#define N_NODES 50000
#define E_EDGES 800000
#define R_REL   6
#define H_HEADS 4
#define DIM     128   // G_DIM == H1 == D_OUT
#define HD      32    // H2, per-head dim
#define EPS_BN  1e-5f

typedef float v2f __attribute__((ext_vector_type(2)));
typedef float v8f __attribute__((ext_vector_type(8)));

// ---------------------------------------------------------------------------
// WMMA fp32 GEMM:  C[M x 128] (+)= A[M x 128] @ B[128 x 128] (+ bias) (relu)
// One block = 256 threads = 8 waves. Block stages a 16-row strip of A in LDS
// (padded stride 132 -> conflict-free), wave w owns columns [16w, 16w+16).
// K consumed 4 at a time with V_WMMA_F32_16X16X4_F32.
//
// Fragment layouts per CDNA5 ISA 7.12.2 (wave32):
//   A 16x4 : lane L -> row = L%16 ; a.x = A[row][k0+2*(L/16)], a.y = +1
//   B 4x16 : lane L -> col = L%16 ; b.x = B[k0+2*(L/16)][col], b.y = +1
//   C 16x16: VGPR i, lane L -> row = i + 8*(L/16), col = L%16
//
// ACC / RELU / BIAS are compile-time so the epilogue is branch-free
// (one clause of 8 loads when ACC, fma/max, one clause of 8 stores).
// ---------------------------------------------------------------------------
template <int ACC, int RELU, int BIAS>
__global__ __launch_bounds__(256)
void gemm128_wmma(const float* __restrict__ A, const float* __restrict__ B,
                  float* __restrict__ C, const float* __restrict__ bias)
{
    __shared__ float As[16][132];
    const int row0 = blockIdx.x * 16;
    const int tid  = threadIdx.x;

    // cooperative load of the 16x128 A strip
    for (int i = tid; i < 16 * 128; i += 256) {
        int r = i >> 7, k = i & 127;
        As[r][k] = A[(size_t)(row0 + r) * DIM + k];
    }
    __syncthreads();

    const int wave = tid >> 5;             // 0..7 -> column tile
    const int lane = tid & 31;
    const int l16  = lane & 15;
    const int hi   = lane >> 4;
    const int col  = wave * 16 + l16;

    v8f acc = {};
    #pragma unroll 4
    for (int k0 = 0; k0 < 128; k0 += 4) {
        const int ka = k0 + 2 * hi;
        v2f a, b;
        a.x = As[l16][ka];
        a.y = As[l16][ka + 1];
        b.x = B[(size_t)ka * DIM + col];
        b.y = B[(size_t)(ka + 1) * DIM + col];
        acc = __builtin_amdgcn_wmma_f32_16x16x4_f32(
            /*neg_a=*/false, a, /*neg_b=*/false, b,
            /*c_mod=*/(short)0, acc, /*reuse_a=*/false, /*reuse_b=*/false);
    }

    float* cp = C + (size_t)(row0 + 8 * hi) * DIM + col;   // rows step by DIM

    float bv = 0.0f;
    if (BIAS) bv = bias[col];

    float cin[8];
    if (ACC) {
        #pragma unroll
        for (int i = 0; i < 8; ++i) cin[i] = cp[(size_t)i * DIM];
    }

    #pragma unroll
    for (int i = 0; i < 8; ++i) {
        float val = acc[i] + bv;
        if (ACC)  val += cin[i];
        if (RELU) val = fmaxf(val, 0.0f);
        cp[(size_t)i * DIM] = val;
    }
}

// ---------------------------------------------------------------------------
// Edge-count per (relation, dst) segment
// ---------------------------------------------------------------------------
__global__ void count_kernel(const int* __restrict__ ei, const int* __restrict__ et,
                             float* __restrict__ cnt)
{
    int e = blockIdx.x * blockDim.x + threadIdx.x;
    if (e >= E_EDGES) return;
    int dst = ei[E_EDGES + e];
    int r   = et[e];
    atomicAdd(&cnt[(size_t)r * N_NODES + dst], 1.0f);
}

// ---------------------------------------------------------------------------
// Weighted scatter for one relation: acc[dst] += x[src] / max(cnt,1)
// 32 lanes per edge, float4 per lane.
// ---------------------------------------------------------------------------
__global__ void rgcn_scatter(const int* __restrict__ ei, const int* __restrict__ et,
                             const float* __restrict__ cnt, const float* __restrict__ x,
                             float* __restrict__ acc, int rel)
{
    int t = blockIdx.x * blockDim.x + threadIdx.x;
    int e = t >> 5;
    if (e >= E_EDGES) return;
    if (et[e] != rel) return;
    int lane = t & 31;
    int src = ei[e];
    int dst = ei[E_EDGES + e];
    float w = 1.0f / fmaxf(cnt[(size_t)rel * N_NODES + dst], 1.0f);
    const float4 xv = *(const float4*)(x + (size_t)src * DIM + lane * 4);
    float* ap = acc + (size_t)dst * DIM + lane * 4;
    atomicAdd(ap + 0, xv.x * w);
    atomicAdd(ap + 1, xv.y * w);
    atomicAdd(ap + 2, xv.z * w);
    atomicAdd(ap + 3, xv.w * w);
}

__global__ void fill_neginf(float* __restrict__ p, int n)
{
    int i = blockIdx.x * blockDim.x + threadIdx.x;
    if (i < n) p[i] = -INFINITY;
}

__device__ __forceinline__ void atomicMaxF(float* addr, float val)
{
    unsigned int* ua = (unsigned int*)addr;
    unsigned int old = *ua;
    while (__uint_as_float(old) < val) {
        unsigned int assumed = old;
        old = atomicCAS(ua, assumed, __float_as_uint(val));
        if (old == assumed) break;
    }
}

// ---------------------------------------------------------------------------
// logits[e,h] = dot(q[dst,h], k[src,h]) / sqrt(HD);  m[dst,h] = segmax
// ---------------------------------------------------------------------------
__global__ void logits_kernel(const int* __restrict__ ei, const float* __restrict__ q,
                              const float* __restrict__ k, float* __restrict__ logits,
                              float* __restrict__ m)
{
    int t = blockIdx.x * blockDim.x + threadIdx.x;
    if (t >= E_EDGES * H_HEADS) return;
    int e = t >> 2, head = t & 3;
    int src = ei[e];
    int dst = ei[E_EDGES + e];
    const float* qp = q + (size_t)dst * DIM + head * HD;
    const float* kp = k + (size_t)src * DIM + head * HD;
    float s = 0.0f;
    #pragma unroll
    for (int i = 0; i < HD; ++i) s += qp[i] * kp[i];
    s *= 0.17677669529663687f;   // 1/sqrt(32)
    logits[t] = s;
    atomicMaxF(&m[(size_t)dst * H_HEADS + head], s);
}

__global__ void fixup_m(float* __restrict__ m, int n)
{
    int i = blockIdx.x * blockDim.x + threadIdx.x;
    if (i < n && !isfinite(m[i])) m[i] = 0.0f;
}

// e = exp(logit - m[dst]); z[dst] += e   (e overwrites the logits buffer)
__global__ void exp_kernel(const int* __restrict__ ei, float* __restrict__ logits,
                           const float* __restrict__ m, float* __restrict__ z)
{
    int t = blockIdx.x * blockDim.x + threadIdx.x;
    if (t >= E_EDGES * H_HEADS) return;
    int e = t >> 2, head = t & 3;
    int dst = ei[E_EDGES + e];
    float ev = expf(logits[t] - m[(size_t)dst * H_HEADS + head]);
    logits[t] = ev;
    atomicAdd(&z[(size_t)dst * H_HEADS + head], ev);
}

// out[dst] += alpha * v[src]; 32 lanes/edge, float4/lane, head = lane/8
__global__ void attn_scatter(const int* __restrict__ ei, const float* __restrict__ eexp,
                             const float* __restrict__ z, const float* __restrict__ v,
                             float* __restrict__ out)
{
    int t = blockIdx.x * blockDim.x + threadIdx.x;
    int e = t >> 5;
    if (e >= E_EDGES) return;
    int lane = t & 31;
    int head = lane >> 3;
    int src = ei[e];
    int dst = ei[E_EDGES + e];
    float alpha = eexp[(size_t)e * H_HEADS + head] /
                  fmaxf(z[(size_t)dst * H_HEADS + head], 1e-16f);
    const float4 vv = *(const float4*)(v + (size_t)src * DIM + lane * 4);
    float* op = out + (size_t)dst * DIM + lane * 4;
    atomicAdd(op + 0, alpha * vv.x);
    atomicAdd(op + 1, alpha * vv.y);
    atomicAdd(op + 2, alpha * vv.z);
    atomicAdd(op + 3, alpha * vv.w);
}

// ReLU in place + per-column sum / sumsq (block = 128 cols x 256 rows)
__global__ __launch_bounds__(128)
void relu_stats(float* __restrict__ y, float* __restrict__ ssum, float* __restrict__ ssq)
{
    int c  = threadIdx.x;
    int r0 = blockIdx.x * 256;
    int r1 = min(r0 + 256, N_NODES);
    float s = 0.0f, sq = 0.0f;
    for (int r = r0; r < r1; ++r) {
        float* p = y + (size_t)r * DIM + c;
        float val = fmaxf(*p, 0.0f);
        *p = val;
        s += val;
        sq += val * val;
    }
    atomicAdd(&ssum[c], s);
    atomicAdd(&ssq[c], sq);
}

__global__ void bn_normalize(float* __restrict__ y, const float* __restrict__ ssum,
                             const float* __restrict__ ssq, const float* __restrict__ gamma,
                             const float* __restrict__ beta)
{
    int i = blockIdx.x * blockDim.x + threadIdx.x;
    if (i >= N_NODES * DIM) return;
    int c = i & (DIM - 1);
    float mu  = ssum[c] * (1.0f / N_NODES);
    float var = fmaxf(ssq[c] * (1.0f / N_NODES) - mu * mu, 0.0f);
    y[i] = (y[i] - mu) * rsqrtf(var + EPS_BN) * gamma[c] + beta[c];
}

// ---------------------------------------------------------------------------
// Host launcher
// ---------------------------------------------------------------------------
extern "C" void kernel_launch(void* const* d_in, const int* in_sizes, int n_in,
                              void* d_out, int out_size, void* d_ws, size_t ws_size,
                              hipStream_t stream)
{
    const float* x        = (const float*)d_in[0];
    const int*   ei       = (const int*)  d_in[1];   // (2,E): [0:E]=src, [E:2E]=dst
    const int*   et       = (const int*)  d_in[2];
    const float* rgcn_w   = (const float*)d_in[3];   // (R,128,128)
    const float* rgcn_root= (const float*)d_in[4];
    const float* rgcn_b   = (const float*)d_in[5];
    const float* wq = (const float*)d_in[6];  const float* bq = (const float*)d_in[7];
    const float* wk = (const float*)d_in[8];  const float* bk = (const float*)d_in[9];
    const float* wv = (const float*)d_in[10]; const float* bv = (const float*)d_in[11];
    const float* wsk= (const float*)d_in[12]; const float* bsk= (const float*)d_in[13];
    const float* wrs= (const float*)d_in[14]; const float* brs= (const float*)d_in[15];
    const float* gamma = (const float*)d_in[16];
    const float* beta  = (const float*)d_in[17];
    float* out = (float*)d_out;

    // workspace carve-up (floats), ~144 MB total
    float* ws = (float*)d_ws;
    size_t o = 0;
    float* h      = ws + o; o += (size_t)N_NODES * DIM;
    float* acc    = ws + o; o += (size_t)N_NODES * DIM;   // reused per relation
    float* qbuf   = ws + o; o += (size_t)N_NODES * DIM;
    float* kbuf   = ws + o; o += (size_t)N_NODES * DIM;
    float* vbuf   = ws + o; o += (size_t)N_NODES * DIM;
    float* cnt    = ws + o; o += (size_t)R_REL * N_NODES;
    float* logits = ws + o; o += (size_t)E_EDGES * H_HEADS;   // later holds e
    float* mmax   = ws + o; o += (size_t)N_NODES * H_HEADS;
    float* zsum   = ws + o; o += (size_t)N_NODES * H_HEADS;
    float* ssum   = ws + o; o += DIM;
    float* ssq    = ws + o; o += DIM;

    const int rowTiles = N_NODES / 16;   // 50000/16 = 3125 exactly

    // ---- init (every call: deterministic) ----
    hipMemsetAsync(h,    0, (size_t)N_NODES * DIM * 4, stream);
    hipMemsetAsync(cnt,  0, (size_t)R_REL * N_NODES * 4, stream);
    hipMemsetAsync(zsum, 0, (size_t)N_NODES * H_HEADS * 4, stream);
    hipMemsetAsync(ssum, 0, DIM * 4, stream);
    hipMemsetAsync(ssq,  0, DIM * 4, stream);
    fill_neginf<<<(N_NODES * H_HEADS + 255) / 256, 256, 0, stream>>>(mmax, N_NODES * H_HEADS);

    // ---- RGCN: counts, then per-relation weighted scatter + GEMM accumulate ----
    count_kernel<<<(E_EDGES + 255) / 256, 256, 0, stream>>>(ei, et, cnt);
    for (int r = 0; r < R_REL; ++r) {
        hipMemsetAsync(acc, 0, (size_t)N_NODES * DIM * 4, stream);
        rgcn_scatter<<<(E_EDGES * 32 + 255) / 256, 256, 0, stream>>>(ei, et, cnt, x, acc, r);
        gemm128_wmma<1, 0, 0><<<rowTiles, 256, 0, stream>>>(
            acc, rgcn_w + (size_t)r * DIM * DIM, h, nullptr);
    }
    // root transform + bias + relu:  h = relu(h + x @ root + b)
    gemm128_wmma<1, 1, 1><<<rowTiles, 256, 0, stream>>>(x, rgcn_root, h, rgcn_b);

    // ---- q / k / v projections ----
    gemm128_wmma<0, 0, 1><<<rowTiles, 256, 0, stream>>>(h, wq, qbuf, bq);
    gemm128_wmma<0, 0, 1><<<rowTiles, 256, 0, stream>>>(h, wk, kbuf, bk);
    gemm128_wmma<0, 0, 1><<<rowTiles, 256, 0, stream>>>(h, wv, vbuf, bv);

    // ---- segment softmax attention ----
    logits_kernel<<<(E_EDGES * H_HEADS + 255) / 256, 256, 0, stream>>>(ei, qbuf, kbuf, logits, mmax);
    fixup_m<<<(N_NODES * H_HEADS + 255) / 256, 256, 0, stream>>>(mmax, N_NODES * H_HEADS);
    exp_kernel<<<(E_EDGES * H_HEADS + 255) / 256, 256, 0, stream>>>(ei, logits, mmax, zsum);

    // ---- skip + residual GEMMs initialize d_out, then attention scatters in ----
    gemm128_wmma<0, 0, 1><<<rowTiles, 256, 0, stream>>>(h, wsk, out, bsk);   // out = h@wskip + bskip
    gemm128_wmma<1, 0, 1><<<rowTiles, 256, 0, stream>>>(h, wrs, out, brs);   // out += h@wres + bres
    attn_scatter<<<(E_EDGES * 32 + 255) / 256, 256, 0, stream>>>(ei, logits, zsum, vbuf, out);

    // ---- relu + batchnorm (training-mode batch stats) ----
    relu_stats<<<(N_NODES + 255) / 256, 128, 0, stream>>>(out, ssum, ssq);
    bn_normalize<<<(N_NODES * DIM + 255) / 256, 256, 0, stream>>>(out, ssum, ssq, gamma, beta);
}